// SparseConvCausalAttention_27779848471407
// MI455X (gfx1250) — compile-verified
//
#include <hip/hip_runtime.h>
#include <hip/hip_bf16.h>

typedef float v2f __attribute__((ext_vector_type(2)));
typedef float v8f __attribute__((ext_vector_type(8)));

#define SEQP 1280
#define NTXT 256
#define NIMG 1024
#define DIMC 512
#define NQKV 1536
#define HEADS 8
#define DHC 64
#define NEG_BIG (-3.0e38f)

// ---- WMMA f32 16x16x4: D = A(16x4) * B(4x16) + C --------------------------
__device__ __forceinline__ v8f wmma4(v2f a, v2f b, v8f c) {
  return __builtin_amdgcn_wmma_f32_16x16x4_f32(
      /*neg_a=*/false, a, /*neg_b=*/false, b,
      /*c_mod=*/(short)0, c, /*reuse_a=*/false, /*reuse_b=*/false);
}

// reduce within each 16-lane half of the wave32 (xor masks 1,2,4,8 stay in-half)
__device__ __forceinline__ float half_max(float v) {
  v = fmaxf(v, __shfl_xor(v, 1));
  v = fmaxf(v, __shfl_xor(v, 2));
  v = fmaxf(v, __shfl_xor(v, 4));
  v = fmaxf(v, __shfl_xor(v, 8));
  return v;
}
__device__ __forceinline__ float half_sum(float v) {
  v += __shfl_xor(v, 1);
  v += __shfl_xor(v, 2);
  v += __shfl_xor(v, 4);
  v += __shfl_xor(v, 8);
  return v;
}

__device__ __forceinline__ float dot64(const float* __restrict__ a,
                                       const float* __restrict__ b) {
  float s = 0.f;
#pragma unroll
  for (int c = 0; c < DHC; c += 4) {
    float4 av = *(const float4*)(a + c);
    float4 bv = *(const float4*)(b + c);
    s = fmaf(av.x, bv.x, s); s = fmaf(av.y, bv.y, s);
    s = fmaf(av.z, bv.z, s); s = fmaf(av.w, bv.w, s);
  }
  return s;
}

// ---------------------------------------------------------------------------
// Kernel 1: qkv = pad(x) @ W_qkv, scattered into per-head q/k/v buffers
// (q pre-scaled by DH^-0.5). Wave computes a 16x64 tile; block = 8 waves.
// Pad row handled branch-free: clamp address + multiply fragment by 0/1
// selector, so the inner loop is a pure load-clause + 4 WMMAs.
// ---------------------------------------------------------------------------
__global__ __launch_bounds__(256) void qkv_gemm(const float* __restrict__ x,
                                                const float* __restrict__ Wqkv,
                                                float* __restrict__ qh,
                                                float* __restrict__ kh,
                                                float* __restrict__ vh) {
  const int lane = threadIdx.x & 31;
  const int wave = threadIdx.x >> 5;
  const int l16 = lane & 15, ha = lane >> 4;
  const int mt = blockIdx.x * 8 + wave;   // 0..319 m-tile (16 rows each)
  const int n0 = blockIdx.y * 64;         // output column strip

  const int arow = mt * 16 + l16;
  const int ab = arow / SEQP;
  const int an = arow % SEQP;
  const bool apad = (an == 1279);          // zero-padded row
  const float zsel = apad ? 0.f : 1.f;     // branch-free zero selector
  const int ansafe = apad ? 0 : an;        // keep address in-bounds
  const float* xrow = x + ((size_t)ab * 1279 + ansafe) * DIMC + 2 * ha;

  v8f acc[4] = {};
  for (int kk = 0; kk < DIMC; kk += 4) {
    v2f a;
    { const float* p = xrow + kk; a[0] = p[0] * zsel; a[1] = p[1] * zsel; }
    const int kb = kk + 2 * ha;
    const float* wb = Wqkv + (size_t)kb * NQKV + n0 + l16;
#pragma unroll
    for (int t = 0; t < 4; ++t) {
      v2f b; b[0] = wb[t * 16]; b[1] = wb[NQKV + t * 16];
      acc[t] = wmma4(a, b, acc[t]);
    }
  }

  // 64-col strip aligned to 64 => single section & single head per strip
  const int sec = n0 / DIMC;                 // 0=q 1=k 2=v
  const int h   = (n0 % DIMC) / DHC;
  float* dstbuf = (sec == 0) ? qh : (sec == 1) ? kh : vh;
  const float scl = (sec == 0) ? 0.125f : 1.0f;
  const int bb = (mt * 16) / SEQP, nbase = (mt * 16) % SEQP;
#pragma unroll
  for (int r = 0; r < 8; ++r) {
    const int n = nbase + r + 8 * ha;
    float* drow = dstbuf + (((size_t)bb * HEADS + h) * SEQP + n) * DHC;
#pragma unroll
    for (int t = 0; t < 4; ++t) drow[t * 16 + l16] = acc[t][r] * scl;
  }
}

// ---------------------------------------------------------------------------
// Kernel 2: causal text attention, flash-style. One wave per (head, 16-query
// tile). Streams key tiles, online softmax in C-fragment layout, P staged
// through LDS to feed the P@V WMMA in A-layout.
// ---------------------------------------------------------------------------
__global__ __launch_bounds__(32) void attn_text(const float* __restrict__ qh,
                                                const float* __restrict__ kh,
                                                const float* __restrict__ vh,
                                                float* __restrict__ oh) {
  __shared__ float sP[256];
  const int lane = threadIdx.x & 31;
  const int l16 = lane & 15, ha = lane >> 4;
  const int bh = blockIdx.x;          // 0..31
  const int q0 = blockIdx.y * 16;     // 0..255
  const float* qbase = qh + (size_t)bh * SEQP * DHC;
  const float* kbase = kh + (size_t)bh * SEQP * DHC;
  const float* vbase = vh + (size_t)bh * SEQP * DHC;

  v2f qa[16];
  {
    const float* qrow = qbase + (size_t)(q0 + l16) * DHC + 2 * ha;
#pragma unroll
    for (int s = 0; s < 16; ++s) { qa[s][0] = qrow[s * 4]; qa[s][1] = qrow[s * 4 + 1]; }
  }
  float mM[8], mL[8];
#pragma unroll
  for (int r = 0; r < 8; ++r) { mM[r] = NEG_BIG; mL[r] = 0.f; }
  v8f acc[4] = {};

  for (int j0 = 0; j0 <= q0; j0 += 16) {
    v8f S = {};
    const float* krow = kbase + (size_t)(j0 + l16) * DHC + 2 * ha;
#pragma unroll
    for (int s = 0; s < 16; ++s) {
      v2f kb; kb[0] = krow[s * 4]; kb[1] = krow[s * 4 + 1];
      S = wmma4(qa[s], kb, S);
    }
    const bool diag = (j0 == q0);
#pragma unroll
    for (int r = 0; r < 8; ++r) {
      float s = S[r];
      if (diag && (l16 > r + 8 * ha)) s = NEG_BIG;   // causal: key > query
      const float tm = half_max(s);
      const float nm = fmaxf(mM[r], tm);
      const float p  = __expf(s - nm);
      const float ps = half_sum(p);
      const float corr = __expf(mM[r] - nm);
      mL[r] = mL[r] * corr + ps;
      mM[r] = nm;
#pragma unroll
      for (int t = 0; t < 4; ++t) acc[t][r] *= corr;
      sP[(r + 8 * ha) * 16 + l16] = p;
    }
    __syncthreads();
#pragma unroll
    for (int s4 = 0; s4 < 4; ++s4) {
      const int kbi = s4 * 4 + 2 * ha;
      v2f pa; pa[0] = sP[l16 * 16 + kbi]; pa[1] = sP[l16 * 16 + kbi + 1];
      const float* v0 = vbase + (size_t)(j0 + kbi) * DHC + l16;
#pragma unroll
      for (int t = 0; t < 4; ++t) {
        v2f vb; vb[0] = v0[t * 16]; vb[1] = v0[DHC + t * 16];
        acc[t] = wmma4(pa, vb, acc[t]);
      }
    }
    __syncthreads();
  }

  float* orow0 = oh + (size_t)bh * SEQP * DHC;
#pragma unroll
  for (int r = 0; r < 8; ++r) {
    const float inv = 1.0f / mL[r];
    float* orow = orow0 + (size_t)(q0 + r + 8 * ha) * DHC;
#pragma unroll
    for (int t = 0; t < 4; ++t) orow[t * 16 + l16] = acc[t][r] * inv;
  }
}

// ---------------------------------------------------------------------------
// Kernel 3: image attention. Phase 1: 256 text keys via WMMA flash loop with
// text mask. Phase 2: 7x7 causal conv-window keys per query, per-lane VALU,
// merged into the same online-softmax state via LDS handoff.
// ---------------------------------------------------------------------------
__global__ __launch_bounds__(32) void attn_img(const float* __restrict__ qh,
                                               const float* __restrict__ kh,
                                               const float* __restrict__ vh,
                                               const unsigned char* __restrict__ mask,
                                               float* __restrict__ oh) {
  __shared__ float sP[256];
  __shared__ float sO[16 * DHC];
  __shared__ float sM[16];
  __shared__ float sL[16];
  const int lane = threadIdx.x & 31;
  const int l16 = lane & 15, ha = lane >> 4;
  const int bh = blockIdx.x;         // 0..31
  const int qt = blockIdx.y;         // 0..63
  const int q0 = qt * 16;            // image query base (0..1023)
  const int b = bh >> 3;
  const float* qbase = qh + (size_t)bh * SEQP * DHC;
  const float* kbase = kh + (size_t)bh * SEQP * DHC;
  const float* vbase = vh + (size_t)bh * SEQP * DHC;

  // ---- phase 1: text keys --------------------------------------------------
  v2f qa[16];
  {
    const float* qrow = qbase + (size_t)(NTXT + q0 + l16) * DHC + 2 * ha;
#pragma unroll
    for (int s = 0; s < 16; ++s) { qa[s][0] = qrow[s * 4]; qa[s][1] = qrow[s * 4 + 1]; }
  }
  float mM[8], mL[8];
#pragma unroll
  for (int r = 0; r < 8; ++r) { mM[r] = NEG_BIG; mL[r] = 0.f; }
  v8f acc[4] = {};

  for (int j0 = 0; j0 < NTXT; j0 += 16) {
    const bool keyok = mask[b * NTXT + j0 + l16] != 0;
    v8f S = {};
    const float* krow = kbase + (size_t)(j0 + l16) * DHC + 2 * ha;
#pragma unroll
    for (int s = 0; s < 16; ++s) {
      v2f kb; kb[0] = krow[s * 4]; kb[1] = krow[s * 4 + 1];
      S = wmma4(qa[s], kb, S);
    }
#pragma unroll
    for (int r = 0; r < 8; ++r) {
      float s = keyok ? S[r] : NEG_BIG;
      const float tm = half_max(s);
      const float nm = fmaxf(mM[r], tm);
      const float p  = __expf(s - nm);
      const float ps = half_sum(p);
      const float corr = __expf(mM[r] - nm);
      mL[r] = mL[r] * corr + ps;
      mM[r] = nm;
#pragma unroll
      for (int t = 0; t < 4; ++t) acc[t][r] *= corr;
      sP[(r + 8 * ha) * 16 + l16] = p;
    }
    __syncthreads();
#pragma unroll
    for (int s4 = 0; s4 < 4; ++s4) {
      const int kbi = s4 * 4 + 2 * ha;
      v2f pa; pa[0] = sP[l16 * 16 + kbi]; pa[1] = sP[l16 * 16 + kbi + 1];
      const float* v0 = vbase + (size_t)(j0 + kbi) * DHC + l16;
#pragma unroll
      for (int t = 0; t < 4; ++t) {
        v2f vb; vb[0] = v0[t * 16]; vb[1] = v0[DHC + t * 16];
        acc[t] = wmma4(pa, vb, acc[t]);
      }
    }
    __syncthreads();
  }

  // ---- handoff: C-fragment -> row-major LDS --------------------------------
#pragma unroll
  for (int r = 0; r < 8; ++r) {
#pragma unroll
    for (int t = 0; t < 4; ++t) sO[(r + 8 * ha) * DHC + t * 16 + l16] = acc[t][r];
  }
  if (l16 == 0) {
#pragma unroll
    for (int r = 0; r < 8; ++r) { sM[r + 8 * ha] = mM[r]; sL[r + 8 * ha] = mL[r]; }
  }
  __syncthreads();

  // ---- phase 2: 7x7 window keys, row-per-lane (halves split dims) ----------
  const int row = l16;
  const int qi = q0 + row;
  const int y = qi >> 5, xx = qi & 31;
  const int dlo = ha * 32;
  const float m1 = sM[row];
  const float l1 = sL[row];
  const float* qr = qbase + (size_t)(NTXT + qi) * DHC;

  float wmax = NEG_BIG;
  for (int w = 0; w < 49; ++w) {
    const int i = w / 7, j = w - i * 7;
    const int ky = y + i - 3, kx = xx + j - 3;
    const int kidx = ky * 32 + kx;
    if ((unsigned)ky < 32u && (unsigned)kx < 32u && kidx <= qi) {
      const float s = dot64(qr, kbase + (size_t)(NTXT + kidx) * DHC);
      wmax = fmaxf(wmax, s);
    }
  }
  const float m2 = fmaxf(m1, wmax);
  const float c1 = __expf(m1 - m2);
  float l2 = l1 * c1;
  float O[32];
#pragma unroll
  for (int i = 0; i < 32; ++i) O[i] = sO[row * DHC + dlo + i] * c1;

  for (int w = 0; w < 49; ++w) {
    const int i = w / 7, j = w - i * 7;
    const int ky = y + i - 3, kx = xx + j - 3;
    const int kidx = ky * 32 + kx;
    if ((unsigned)ky < 32u && (unsigned)kx < 32u && kidx <= qi) {
      const float s = dot64(qr, kbase + (size_t)(NTXT + kidx) * DHC);
      const float p = __expf(s - m2);
      l2 += p;
      const float* vr = vbase + (size_t)(NTXT + kidx) * DHC + dlo;
#pragma unroll
      for (int c = 0; c < 32; c += 4) {
        float4 vv = *(const float4*)(vr + c);
        O[c + 0] = fmaf(p, vv.x, O[c + 0]);
        O[c + 1] = fmaf(p, vv.y, O[c + 1]);
        O[c + 2] = fmaf(p, vv.z, O[c + 2]);
        O[c + 3] = fmaf(p, vv.w, O[c + 3]);
      }
    }
  }
  const float inv = 1.0f / l2;
  float* orow = oh + ((size_t)bh * SEQP + NTXT + qi) * DHC + dlo;
#pragma unroll
  for (int c = 0; c < 32; c += 4) {
    float4 ov = { O[c] * inv, O[c + 1] * inv, O[c + 2] * inv, O[c + 3] * inv };
    *(float4*)(orow + c) = ov;
  }
}

// ---------------------------------------------------------------------------
// Kernel 4: out = merge_heads(oh) @ W_out + b_out, dropping the pad row.
// ---------------------------------------------------------------------------
__global__ __launch_bounds__(256) void out_gemm(const float* __restrict__ oh,
                                                const float* __restrict__ Wout,
                                                const float* __restrict__ bout,
                                                float* __restrict__ out) {
  const int lane = threadIdx.x & 31;
  const int wave = threadIdx.x >> 5;
  const int l16 = lane & 15, ha = lane >> 4;
  const int mt = blockIdx.x * 8 + wave;   // 0..319
  const int n0 = blockIdx.y * 64;         // 0..511

  const int arow = mt * 16 + l16;
  const int ab = arow / SEQP;
  const int an = arow % SEQP;

  v8f acc[4] = {};
  for (int kk = 0; kk < DIMC; kk += 4) {
    const int ka = kk + 2 * ha;
    const int h = ka / DHC, d = ka % DHC;   // ka, ka+1 stay inside one head
    const float* ar = oh + (((size_t)ab * HEADS + h) * SEQP + an) * DHC + d;
    v2f a; a[0] = ar[0]; a[1] = ar[1];
    const float* wb = Wout + (size_t)ka * DIMC + n0 + l16;
#pragma unroll
    for (int t = 0; t < 4; ++t) {
      v2f b; b[0] = wb[t * 16]; b[1] = wb[DIMC + t * 16];
      acc[t] = wmma4(a, b, acc[t]);
    }
  }

  const int bb = (mt * 16) / SEQP, nbase = (mt * 16) % SEQP;
#pragma unroll
  for (int r = 0; r < 8; ++r) {
    const int n = nbase + r + 8 * ha;
    if (n < 1279) {
      float* orow = out + ((size_t)bb * 1279 + n) * DIMC + n0;
#pragma unroll
      for (int t = 0; t < 4; ++t)
        orow[t * 16 + l16] = acc[t][r] + bout[n0 + t * 16 + l16];
    }
  }
}

// ---------------------------------------------------------------------------
extern "C" void kernel_launch(void* const* d_in, const int* in_sizes, int n_in,
                              void* d_out, int out_size, void* d_ws, size_t ws_size,
                              hipStream_t stream) {
  (void)in_sizes; (void)n_in; (void)out_size; (void)ws_size;
  const float* x            = (const float*)d_in[0];
  const unsigned char* mask = (const unsigned char*)d_in[1];
  const float* Wqkv         = (const float*)d_in[2];
  const float* Wout         = (const float*)d_in[3];
  const float* bout         = (const float*)d_in[4];

  float* ws = (float*)d_ws;
  const size_t seg = (size_t)32 * SEQP * DHC;   // 2,621,440 floats per buffer
  float* qh = ws;
  float* kh = ws + seg;
  float* vh = ws + 2 * seg;
  float* vh2 = vh;
  float* oh = ws + 3 * seg;
  (void)vh2;

  qkv_gemm<<<dim3(40, 24), 256, 0, stream>>>(x, Wqkv, qh, kh, vh);
  attn_text<<<dim3(32, 16), 32, 0, stream>>>(qh, kh, vh, oh);
  attn_img<<<dim3(32, 64), 32, 0, stream>>>(qh, kh, vh, mask, oh);
  out_gemm<<<dim3(40, 8), 256, 0, stream>>>(oh, Wout, bout, (float*)d_out);
}